// EigenvectorSimilarity_65446711656861
// MI455X (gfx1250) — compile-verified
//
#include <hip/hip_runtime.h>
#include <math.h>

// ---------------------------------------------------------------------------
// EigenvectorSimilarity for MI455X (gfx1250, wave32, WMMA).
// Pipeline per matrix:
//   1) row norms
//   2) cosine-sim Gram via V_WMMA_F32_16X16X4_F32, threshold -> adjacency
//   3) row-sum degrees, in-place Laplacian L = D - A
//   4) Householder tridiagonalization (host loop; matvec + rank-2 update
//      are device-wide memory-bound kernels; reductions single-block,
//      fixed-order -> deterministic)
//   5) Sturm-sequence bisection: one thread per eigenvalue (ascending),
//      stored descending
// Final: cumulative-energy cutoffs k1,k2, final_k=min, masked sum of squared
// eigenvalue differences -> d_out[0].
// ---------------------------------------------------------------------------

typedef __attribute__((ext_vector_type(2))) float v2f;
typedef __attribute__((ext_vector_type(8))) float v8f;

#define NMAT 4096
#define KDIM 256
#define SIM_EPS 1e-6f
#define SIM_THRESH 0.9f
#define ENERGY_FRAC 0.9f

// ---------------------------- 1) row norms ---------------------------------
__global__ void row_norms_kernel(const float* __restrict__ E,
                                 float* __restrict__ norms, int k) {
  __shared__ float red[256];
  const int row = blockIdx.x;
  const int tid = threadIdx.x;
  float acc = 0.f;
  for (int j = tid; j < k; j += blockDim.x) {
    float x = E[(size_t)row * k + j];
    acc += x * x;
  }
  red[tid] = acc;
  __syncthreads();
  for (int s = 128; s > 0; s >>= 1) {
    if (tid < s) red[tid] += red[tid + s];
    __syncthreads();
  }
  if (tid == 0) norms[row] = sqrtf(red[0]);
}

// ------------- 2) WMMA Gram (E E^T) -> cosine sim -> adjacency -------------
// One wave per 16x16 output tile; K reduced in steps of 4 with
// V_WMMA_F32_16X16X4_F32 (fp32 keeps the >0.9 threshold exact).
__global__ void gram_adj_kernel(const float* __restrict__ E,
                                const float* __restrict__ norms,
                                float* __restrict__ Adj, int n, int k) {
  const int wavesPerBlock = blockDim.x >> 5;
  const int tile = blockIdx.x * wavesPerBlock + (threadIdx.x >> 5);
  const int nt = n >> 4;                       // tiles per dim
  const int tm = (tile / nt) << 4;             // tile row base
  const int tn = (tile % nt) << 4;             // tile col base
  const int lane = threadIdx.x & 31;
  const int r = lane & 15;                     // row-in-tile for operand loads
  const int koff = (lane >> 4) << 1;           // 0 for lanes 0-15, 2 for 16-31

  const float* __restrict__ arow = E + (size_t)(tm + r) * k;   // A: rows of tile M
  const float* __restrict__ brow = E + (size_t)(tn + r) * k;   // B: rows of tile N (E^T cols)

  v8f acc = {0.f, 0.f, 0.f, 0.f, 0.f, 0.f, 0.f, 0.f};
#pragma unroll 8
  for (int k0 = 0; k0 < k; k0 += 4) {
    v2f a = *(const v2f*)(arow + k0 + koff);   // {A[r][k0+koff], A[r][k0+koff+1]}
    v2f b = *(const v2f*)(brow + k0 + koff);
    acc = __builtin_amdgcn_wmma_f32_16x16x4_f32(
        /*neg_a=*/false, a, /*neg_b=*/false, b,
        /*c_mod=*/(short)0, acc, /*reuse_a=*/false, /*reuse_b=*/false);
  }

  // D layout: VGPR j, lanes 0-15 -> (M=j, N=lane), lanes 16-31 -> (M=j+8, N=lane-16)
  const int col = tn + (lane & 15);
  const int mhalf = (lane >> 4) << 3;
  const float ncol = norms[col];
#pragma unroll
  for (int j = 0; j < 8; ++j) {
    const int row = tm + mhalf + j;
    const float denom = fmaxf(norms[row] * ncol, SIM_EPS);
    const float sim = acc[j] / denom;
    Adj[(size_t)row * n + col] = (sim > SIM_THRESH) ? 1.0f : 0.0f;
  }
}

// ---------------- 3) in-place Laplacian: L = diag(deg) - A -----------------
__global__ void laplacian_rows_kernel(float* __restrict__ L, int n) {
  __shared__ float red[256];
  __shared__ float s_deg;
  const int row = blockIdx.x;
  const int tid = threadIdx.x;
  float acc = 0.f;
  for (int j = tid; j < n; j += blockDim.x) acc += L[(size_t)row * n + j];
  red[tid] = acc;
  __syncthreads();
  for (int s = 128; s > 0; s >>= 1) {
    if (tid < s) red[tid] += red[tid + s];
    __syncthreads();
  }
  if (tid == 0) s_deg = red[0];
  __syncthreads();
  const float deg = s_deg;
  for (int j = tid; j < n; j += blockDim.x) {
    const float a = L[(size_t)row * n + j];
    L[(size_t)row * n + j] = (j == row) ? (deg - a) : (-a);
  }
}

// ---------------- 4) Householder tridiagonalization step -------------------
// prepare: build unit Householder vector v from column k below the diagonal.
__global__ void hh_prepare_kernel(float* __restrict__ L, int n, int k,
                                  float* __restrict__ v,
                                  float* __restrict__ dvec,
                                  float* __restrict__ evec) {
  __shared__ float red[512];
  __shared__ float s_inv, s_alpha;
  const int m = n - k - 1;
  const int tid = threadIdx.x;
  float sq = 0.f;
  for (int i = tid; i < m; i += blockDim.x) {
    const float x = L[(size_t)(k + 1 + i) * n + k];
    sq += x * x;
  }
  red[tid] = sq;
  __syncthreads();
  for (int s = 256; s > 0; s >>= 1) {
    if (tid < s) red[tid] += red[tid + s];
    __syncthreads();
  }
  if (tid == 0) {
    const float total = red[0];
    const float x0 = L[(size_t)(k + 1) * n + k];
    const float xn = sqrtf(total);
    const float alpha = (x0 >= 0.f) ? -xn : xn;
    dvec[k] = L[(size_t)k * n + k];
    evec[k] = alpha;
    const float vn2 = 2.f * (total - alpha * x0);  // ||x - alpha e1||^2
    s_inv = (vn2 > 1e-30f) ? rsqrtf(vn2) : 0.f;    // 0 => no-op step
    s_alpha = alpha;
  }
  __syncthreads();
  const float inv = s_inv;
  const float alpha = s_alpha;
  for (int i = tid; i < m; i += blockDim.x) {
    const float x = L[(size_t)(k + 1 + i) * n + k];
    v[i] = (x - (i == 0 ? alpha : 0.f)) * inv;
  }
}

// matvec: p = L_trailing * v (one block per trailing row; memory-bound BLAS2)
__global__ void hh_matvec_kernel(const float* __restrict__ L, int n, int k,
                                 const float* __restrict__ v,
                                 float* __restrict__ p) {
  __shared__ float red[256];
  const int m = n - k - 1;
  const int i = blockIdx.x;
  if (i >= m) return;
  const float* __restrict__ row = L + (size_t)(k + 1 + i) * n + (k + 1);
  float acc = 0.f;
  for (int j = threadIdx.x; j < m; j += blockDim.x) acc += row[j] * v[j];
  red[threadIdx.x] = acc;
  __syncthreads();
  for (int s = 128; s > 0; s >>= 1) {
    if (threadIdx.x < s) red[threadIdx.x] += red[threadIdx.x + s];
    __syncthreads();
  }
  if (threadIdx.x == 0) p[i] = red[0];
}

// dot: s = v . p (single block, fixed-order -> deterministic across replays)
__global__ void hh_dot_kernel(const float* __restrict__ v,
                              const float* __restrict__ p, int m,
                              float* __restrict__ scal) {
  __shared__ float red[1024];
  float acc = 0.f;
  for (int i = threadIdx.x; i < m; i += blockDim.x) acc += v[i] * p[i];
  red[threadIdx.x] = acc;
  __syncthreads();
  for (int s = 512; s > 0; s >>= 1) {
    if (threadIdx.x < s) red[threadIdx.x] += red[threadIdx.x + s];
    __syncthreads();
  }
  if (threadIdx.x == 0) scal[1] = red[0];
}

// rank-2 update: L -= v w^T + w v^T, w = 2p - 2(v.p)v
__global__ void hh_update_kernel(float* __restrict__ L, int n, int k,
                                 const float* __restrict__ v,
                                 const float* __restrict__ p,
                                 const float* __restrict__ scal) {
  const int m = n - k - 1;
  const int j = blockIdx.x * 16 + threadIdx.x;
  const int i = blockIdx.y * 16 + threadIdx.y;
  if (i >= m || j >= m) return;
  const float s = scal[1];
  const float vi = v[i], vj = v[j];
  const float wi = 2.f * p[i] - 2.f * s * vi;
  const float wj = 2.f * p[j] - 2.f * s * vj;
  L[(size_t)(k + 1 + i) * n + (k + 1 + j)] -= vi * wj + wi * vj;
}

// pull the last 2x2 block's diag/offdiag out of the reduced matrix
__global__ void extract_tail_kernel(const float* __restrict__ L, int n,
                                    float* __restrict__ dvec,
                                    float* __restrict__ evec) {
  dvec[n - 2] = L[(size_t)(n - 2) * n + (n - 2)];
  dvec[n - 1] = L[(size_t)(n - 1) * n + (n - 1)];
  evec[n - 2] = L[(size_t)(n - 1) * n + (n - 2)];
}

// ---------------- 5) Sturm bisection eigenvalues ---------------------------
__global__ void gersh_bounds_kernel(const float* __restrict__ dvec,
                                    const float* __restrict__ evec, int n,
                                    float* __restrict__ scal) {
  __shared__ float rlo[256], rhi[256];
  float lo = 1e30f, hi = -1e30f;
  for (int i = threadIdx.x; i < n; i += blockDim.x) {
    const float r = (i > 0 ? fabsf(evec[i - 1]) : 0.f) +
                    (i < n - 1 ? fabsf(evec[i]) : 0.f);
    lo = fminf(lo, dvec[i] - r);
    hi = fmaxf(hi, dvec[i] + r);
  }
  rlo[threadIdx.x] = lo;
  rhi[threadIdx.x] = hi;
  __syncthreads();
  for (int s = 128; s > 0; s >>= 1) {
    if (threadIdx.x < s) {
      rlo[threadIdx.x] = fminf(rlo[threadIdx.x], rlo[threadIdx.x + s]);
      rhi[threadIdx.x] = fmaxf(rhi[threadIdx.x], rhi[threadIdx.x + s]);
    }
    __syncthreads();
  }
  if (threadIdx.x == 0) {
    scal[2] = rlo[0] - 1e-3f;
    scal[3] = rhi[0] + 1e-3f;
  }
}

__global__ void sturm_bisect_kernel(const float* __restrict__ dvec,
                                    const float* __restrict__ evec, int n,
                                    const float* __restrict__ scal,
                                    float* __restrict__ eig_desc) {
  const int t = blockIdx.x * blockDim.x + threadIdx.x;  // find t-th smallest
  if (t >= n) return;
  float lo = scal[2], hi = scal[3];
  for (int it = 0; it < 64; ++it) {
    const float mid = 0.5f * (lo + hi);
    int cnt = 0;
    float q = dvec[0] - mid;
    if (q < 0.f) cnt++;
    for (int i = 1; i < n; ++i) {
      const float e = evec[i - 1];
      float denom = q;
      if (fabsf(denom) < 1e-30f) denom = (denom < 0.f) ? -1e-30f : 1e-30f;
      q = dvec[i] - mid - (e * e) / denom;
      if (q < 0.f) cnt++;
    }
    if (cnt > t) hi = mid; else lo = mid;
  }
  eig_desc[n - 1 - t] = 0.5f * (lo + hi);  // store descending
}

// ---------------- final: cutoff indices + masked SSE -----------------------
__global__ void finalize_kernel(const float* __restrict__ eA,
                                const float* __restrict__ eB, int n,
                                float* __restrict__ out) {
  __shared__ float red[1024];
  __shared__ float s_totA, s_totB;
  __shared__ int s_fk;
  const int tid = threadIdx.x;

  float sa = 0.f, sb = 0.f;
  for (int i = tid; i < n; i += blockDim.x) { sa += eA[i]; sb += eB[i]; }

  red[tid] = sa;
  __syncthreads();
  for (int s = 512; s > 0; s >>= 1) {
    if (tid < s) red[tid] += red[tid + s];
    __syncthreads();
  }
  if (tid == 0) s_totA = red[0];
  __syncthreads();

  red[tid] = sb;
  __syncthreads();
  for (int s = 512; s > 0; s >>= 1) {
    if (tid < s) red[tid] += red[tid + s];
    __syncthreads();
  }
  if (tid == 0) s_totB = red[0];
  __syncthreads();

  if (tid == 0) {
    // argmax(cumsum > 0.9*total): first True, else 0 (jnp.argmax semantics)
    float c = 0.f;
    int kA = 0, kB = 0, found;
    found = 0;
    for (int i = 0; i < n; ++i) {
      c += eA[i];
      if (c > ENERGY_FRAC * s_totA) { kA = i; found = 1; break; }
    }
    if (!found) kA = 0;
    c = 0.f; found = 0;
    for (int i = 0; i < n; ++i) {
      c += eB[i];
      if (c > ENERGY_FRAC * s_totB) { kB = i; found = 1; break; }
    }
    if (!found) kB = 0;
    s_fk = kA < kB ? kA : kB;
  }
  __syncthreads();

  const int fk = s_fk;
  float acc = 0.f;
  for (int i = tid; i < n; i += blockDim.x) {
    if (i < fk) {
      const float d = eA[i] - eB[i];
      acc += d * d;
    }
  }
  red[tid] = acc;
  __syncthreads();
  for (int s = 512; s > 0; s >>= 1) {
    if (tid < s) red[tid] += red[tid + s];
    __syncthreads();
  }
  if (tid == 0) out[0] = red[0];
}

// ---------------------------------------------------------------------------
extern "C" void kernel_launch(void* const* d_in, const int* in_sizes, int n_in,
                              void* d_out, int out_size, void* d_ws, size_t ws_size,
                              hipStream_t stream) {
  const int n = NMAT, k = KDIM;
  const float* src = (const float*)d_in[0];
  const float* trg = (const float*)d_in[1];
  float* out = (float*)d_out;

  // Workspace layout (needs (n*n + 7n + 16)*4 B ~= 64.1 MB)
  float* L     = (float*)d_ws;
  float* norms = L + (size_t)n * n;
  float* v     = norms + n;
  float* p     = v + n;
  float* dvec  = p + n;
  float* evec  = dvec + n;
  float* eigA  = evec + n;
  float* eigB  = eigA + n;
  float* scal  = eigB + n;  // [1]=v.p  [2]=lo  [3]=hi

  const float* ins[2] = {src, trg};
  float* eigs[2] = {eigA, eigB};

  for (int mtx = 0; mtx < 2; ++mtx) {
    const float* E = ins[mtx];

    row_norms_kernel<<<n, 256, 0, stream>>>(E, norms, k);

    const int nt = n / 16;                         // 256 tiles per dim
    gram_adj_kernel<<<(nt * nt) / 8, 256, 0, stream>>>(E, norms, L, n, k);

    laplacian_rows_kernel<<<n, 256, 0, stream>>>(L, n);

    for (int kk = 0; kk <= n - 3; ++kk) {
      const int m = n - kk - 1;
      hh_prepare_kernel<<<1, 512, 0, stream>>>(L, n, kk, v, dvec, evec);
      hh_matvec_kernel<<<m, 256, 0, stream>>>(L, n, kk, v, p);
      hh_dot_kernel<<<1, 1024, 0, stream>>>(v, p, m, scal);
      dim3 ug((m + 15) / 16, (m + 15) / 16);
      dim3 ub(16, 16);
      hh_update_kernel<<<ug, ub, 0, stream>>>(L, n, kk, v, p, scal);
    }
    extract_tail_kernel<<<1, 1, 0, stream>>>(L, n, dvec, evec);

    gersh_bounds_kernel<<<1, 256, 0, stream>>>(dvec, evec, n, scal);
    sturm_bisect_kernel<<<n / 256, 256, 0, stream>>>(dvec, evec, n, scal, eigs[mtx]);
  }

  finalize_kernel<<<1, 1024, 0, stream>>>(eigA, eigB, n, out);
}